// ForwardWarp_53472342835318
// MI455X (gfx1250) — compile-verified
//
#include <hip/hip_runtime.h>
#include <hip/hip_bf16.h>
#include <stddef.h>

// Forward warp (bilinear splatting) for MI455X / gfx1250.
// Shapes fixed by the reference: N=4, C=3, H=1080, W=1920.
//
// Strategy:
//  1) zero-fill output with regular-temporal float4 stores -> output (99.5 MB)
//     becomes resident in the 192 MB device L2, so the subsequent atomic
//     scatter RMWs hit in L2.
//  2) one thread per source pixel, 3D grid (no integer divides):
//     grid = (W/128, H, N), block = 128 threads (4 wave32s).
//     Non-temporal loads of flow + 3 channels (streaming data must not evict
//     the L2-resident output), then up to 4 corners x 3 channels hardware
//     float atomics (global_atomic_add_f32, no-return) with OOB corners
//     dropped (reference uses mode='drop').

#define FW_H 1080
#define FW_W 1920
#define FW_C 3
#define FW_HW (FW_H * FW_W)

// Defined FIRST so the disasm snippet shows this kernel's body
// (verify global_atomic_add_f32 + nt loads).
__global__ __launch_bounds__(128) void fwarp_scatter_kernel(
    const float* __restrict__ img,   // [N, C, H, W]
    const float* __restrict__ flo,   // [N, 2, H, W]
    float* __restrict__ out)         // [N, C, H, W]
{
    const int w = blockIdx.x * 128 + threadIdx.x;   // W = 1920 = 15 * 128
    const int h = blockIdx.y;
    const int n = blockIdx.z;
    const int rem = h * FW_W + w;                   // offset within one image plane

    // flo[:,0] = "y" (paired with cols / W), flo[:,1] = "x" (paired with rows / H)
    const size_t flo_base = (size_t)n * 2 * FW_HW + rem;
    const float y = __builtin_nontemporal_load(flo + flo_base);            // channel 0
    const float x = __builtin_nontemporal_load(flo + flo_base + FW_HW);    // channel 1

    const float x1 = floorf(x);
    const float y1 = floorf(y);
    const int  ix1 = (int)x1;
    const int  iy1 = (int)y1;
    const float fx = x - x1;   // wx: dx==0 -> 1-fx, dx==1 -> fx
    const float fy = y - y1;   // wy: dy==0 -> 1-fy, dy==1 -> fy

    // Source pixel channels (NCHW: stride HW between channels), streamed once.
    const size_t img_base = (size_t)n * FW_C * FW_HW + rem;
    const float v0 = __builtin_nontemporal_load(img + img_base);
    const float v1 = __builtin_nontemporal_load(img + img_base + FW_HW);
    const float v2 = __builtin_nontemporal_load(img + img_base + 2 * FW_HW);

    const size_t out_n_base = (size_t)n * FW_C * FW_HW;

    #pragma unroll
    for (int corner = 0; corner < 4; ++corner) {
        const int dx = corner >> 1;       // row direction (checked vs H)
        const int dy = corner & 1;        // col direction (checked vs W)
        const int r  = ix1 + dx + h;      // dest row
        const int c  = iy1 + dy + w;      // dest col
        if ((unsigned)r < (unsigned)FW_H && (unsigned)c < (unsigned)FW_W) {
            const float wx  = dx ? fx : (1.0f - fx);
            const float wy  = dy ? fy : (1.0f - fy);
            const float wgt = wx * wy;
            const size_t o = out_n_base + (size_t)r * FW_W + c;
            // Hardware FP atomics -> global_atomic_add_f32 (no return form,
            // tracked by STOREcnt; RMW resolves in device L2).
            unsafeAtomicAdd(out + o,               v0 * wgt);
            unsafeAtomicAdd(out + o + FW_HW,       v1 * wgt);
            unsafeAtomicAdd(out + o + 2 * FW_HW,   v2 * wgt);
        }
    }
}

__global__ void fwarp_zero_kernel(float4* __restrict__ out4, int n4) {
    int i = blockIdx.x * blockDim.x + threadIdx.x;
    if (i < n4) {
        // Regular-temporal store: leave output lines resident in L2 for the
        // atomic scatter pass.
        out4[i] = make_float4(0.f, 0.f, 0.f, 0.f);
    }
}

extern "C" void kernel_launch(void* const* d_in, const int* in_sizes, int n_in,
                              void* d_out, int out_size, void* d_ws, size_t ws_size,
                              hipStream_t stream) {
    (void)d_ws; (void)ws_size; (void)n_in;

    const float* img = (const float*)d_in[0];   // N*C*H*W floats
    const float* flo = (const float*)d_in[1];   // N*2*H*W floats
    float* out = (float*)d_out;                 // N*C*H*W floats

    const int n_pix = in_sizes[1] / 2;          // N*H*W
    const int N = n_pix / FW_HW;                // 4

    // 1) zero output (out_size = N*C*H*W = 24,883,200; divisible by 4).
    {
        const int n4 = out_size / 4;
        const int threads = 256;
        const int blocks = (n4 + threads - 1) / threads;
        fwarp_zero_kernel<<<blocks, threads, 0, stream>>>((float4*)out, n4);
    }

    // 2) scatter-add splat: 3D grid, no integer divides in-kernel.
    {
        dim3 block(128, 1, 1);                  // 4 wave32s
        dim3 grid(FW_W / 128, FW_H, N);         // (15, 1080, 4)
        fwarp_scatter_kernel<<<grid, block, 0, stream>>>(img, flo, out);
    }
}